// EvoMamba_43696997270091
// MI455X (gfx1250) — compile-verified
//
#include <hip/hip_runtime.h>
#include <hip/hip_bf16.h>
#include <math.h>

// Problem constants (match reference)
#define LYR   24
#define DM    128     // d_model
#define DI    256     // d_inner
#define DST   16      // d_state
#define DCONV 4
#define DTR   8       // dt_rank
#define BB    2
#define SS    512
#define NT    (BB*SS) // 1024 tokens

typedef __attribute__((ext_vector_type(2))) float v2f;
typedef __attribute__((ext_vector_type(8))) float v8f;

// fp32 WMMA: D(16x16,f32) = A(16x4,f32) x B(4x16,f32) + C
__device__ __forceinline__ v8f wmma4(v2f a, v2f b, v8f c) {
  return __builtin_amdgcn_wmma_f32_16x16x4_f32(false, a, false, b, (short)0, c,
                                               false, false);
}

__device__ __forceinline__ float siluf(float x) { return x / (1.f + __expf(-x)); }

// ---------------------------------------------------------------------------
// Generic C[M,N] = A[M,K] * W[N,K]^T   (row-major, K % 4 == 0, N % 64 == 0)
// One wave per 16x64 output tile: blockDim = 32, grid = (M/16, N/64).
// Fragment layout per CDNA5 ISA 7.12.2 (32-bit A 16x4 / B 4x16 / D 16x16).
// ---------------------------------------------------------------------------
template <int K, int N>
__global__ __launch_bounds__(32)
void k_gemm4(const float* __restrict__ A, const float* __restrict__ W,
             float* __restrict__ C) {
  const int m0   = blockIdx.x * 16;
  const int n0   = blockIdx.y * 64;
  const int lane = threadIdx.x;
  const int mr   = lane & 15;
  const int kh   = (lane >> 4) << 1;   // 0 or 2: lane-half selects K pair
  v8f acc0 = {}, acc1 = {}, acc2 = {}, acc3 = {};
  const float* arow = A + (size_t)(m0 + mr) * K + kh;
  const float* wrow = W + (size_t)(n0 + mr) * K + kh;
  for (int k0 = 0; k0 < K; k0 += 4) {
    v2f a, b0, b1, b2, b3;
    a.x  = arow[k0];            a.y  = arow[k0 + 1];
    b0.x = wrow[k0];            b0.y = wrow[k0 + 1];
    b1.x = wrow[16 * K + k0];   b1.y = wrow[16 * K + k0 + 1];
    b2.x = wrow[32 * K + k0];   b2.y = wrow[32 * K + k0 + 1];
    b3.x = wrow[48 * K + k0];   b3.y = wrow[48 * K + k0 + 1];
    acc0 = wmma4(a, b0, acc0);
    acc1 = wmma4(a, b1, acc1);
    acc2 = wmma4(a, b2, acc2);
    acc3 = wmma4(a, b3, acc3);
  }
  const int mo = m0 + ((lane >> 4) << 3);  // D: rows r + 8*(lane>=16)
  const int nc = n0 + (lane & 15);
#pragma unroll
  for (int r = 0; r < 8; ++r) {
    float* crow = C + (size_t)(mo + r) * N + nc;
    crow[0]  = acc0[r];
    crow[16] = acc1[r];
    crow[32] = acc2[r];
    crow[48] = acc3[r];
  }
}

// ---------------------------------------------------------------------------
// lin GEMM (128x128) fused with ReLU + residual add into s, and history write
// ---------------------------------------------------------------------------
__global__ __launch_bounds__(32)
void k_lin_resid(const float* __restrict__ A, const float* __restrict__ W,
                 float* __restrict__ scur, float* __restrict__ hist) {
  const int K = DM, N = DM;
  const int m0   = blockIdx.x * 16;
  const int n0   = blockIdx.y * 64;
  const int lane = threadIdx.x;
  const int mr   = lane & 15;
  const int kh   = (lane >> 4) << 1;
  v8f acc0 = {}, acc1 = {}, acc2 = {}, acc3 = {};
  const float* arow = A + (size_t)(m0 + mr) * K + kh;
  const float* wrow = W + (size_t)(n0 + mr) * K + kh;
  for (int k0 = 0; k0 < K; k0 += 4) {
    v2f a, b0, b1, b2, b3;
    a.x  = arow[k0];            a.y  = arow[k0 + 1];
    b0.x = wrow[k0];            b0.y = wrow[k0 + 1];
    b1.x = wrow[16 * K + k0];   b1.y = wrow[16 * K + k0 + 1];
    b2.x = wrow[32 * K + k0];   b2.y = wrow[32 * K + k0 + 1];
    b3.x = wrow[48 * K + k0];   b3.y = wrow[48 * K + k0 + 1];
    acc0 = wmma4(a, b0, acc0);
    acc1 = wmma4(a, b1, acc1);
    acc2 = wmma4(a, b2, acc2);
    ac3:;
    acc3 = wmma4(a, b3, acc3);
  }
  const int mo = m0 + ((lane >> 4) << 3);
  const int nc = n0 + (lane & 15);
#pragma unroll
  for (int r = 0; r < 8; ++r) {
    size_t base = (size_t)(mo + r) * N + nc;
    float v0 = fmaxf(acc0[r], 0.f) + scur[base];
    float v1 = fmaxf(acc1[r], 0.f) + scur[base + 16];
    float v2 = fmaxf(acc2[r], 0.f) + scur[base + 32];
    float v3 = fmaxf(acc3[r], 0.f) + scur[base + 48];
    scur[base]      = v0; scur[base + 16] = v1;
    scur[base + 32] = v2; scur[base + 48] = v3;
    hist[base]      = v0; hist[base + 16] = v1;
    hist[base + 32] = v2; hist[base + 48] = v3;
  }
}

// ---------------------------------------------------------------------------
// Fused: causal depthwise conv1d + SiLU -> x_proj -> dt_proj + softplus.
// One block (256 threads) per token; xc row staged in LDS for the x_proj dot.
// ---------------------------------------------------------------------------
__global__ __launch_bounds__(DI)
void k_conv_proj(const float* __restrict__ xz, const float* __restrict__ cw,
                 const float* __restrict__ cb, const float* __restrict__ Wx,
                 const float* __restrict__ Wdt, const float* __restrict__ bdt,
                 float* __restrict__ xc, float* __restrict__ dtb,
                 float* __restrict__ Bm, float* __restrict__ Cm) {
  __shared__ float sx[DI];
  __shared__ float sp[DTR + 2 * DST];
  const int t    = blockIdx.x;          // global token (b*SS + s)
  const int sloc = t & (SS - 1);        // position within sequence
  const int e    = threadIdx.x;

  float acc = cb[e];
#pragma unroll
  for (int k = 0; k < DCONV; ++k) {
    int ts = sloc - (DCONV - 1) + k;
    if (ts >= 0)
      acc += cw[e * DCONV + k] * xz[(size_t)(t - (DCONV - 1) + k) * (2 * DI) + e];
  }
  float xv = siluf(acc);
  sx[e] = xv;
  xc[(size_t)t * DI + e] = xv;
  __syncthreads();

  if (e < DTR + 2 * DST) {              // 40 small dot products of length 256
    float p = 0.f;
    const float* wr = Wx + (size_t)e * DI;
    for (int j = 0; j < DI; ++j) p += wr[j] * sx[j];
    sp[e] = p;
    if (e >= DTR && e < DTR + DST)  Bm[(size_t)t * DST + (e - DTR)]       = p;
    if (e >= DTR + DST)             Cm[(size_t)t * DST + (e - DTR - DST)] = p;
  }
  __syncthreads();

  float d = bdt[e];
#pragma unroll
  for (int r = 0; r < DTR; ++r) d += Wdt[e * DTR + r] * sp[r];
  float dt = (d > 20.f) ? d : log1pf(__expf(d));   // softplus
  dtb[(size_t)t * DI + e] = dt;
}

// ---------------------------------------------------------------------------
// Selective-scan: thread = channel e, 16 states in registers, 512 serial steps.
// Fused epilogue: yact = (y + x*D) * silu(z_gate).
// ---------------------------------------------------------------------------
__global__ __launch_bounds__(DI)
void k_scan(const float* __restrict__ xc, const float* __restrict__ dtb,
            const float* __restrict__ Bm, const float* __restrict__ Cm,
            const float* __restrict__ Alog, const float* __restrict__ Dp,
            const float* __restrict__ xz, float* __restrict__ yact) {
  const int b = blockIdx.x;
  const int e = threadIdx.x;
  float Ae[DST], h[DST];
#pragma unroll
  for (int n = 0; n < DST; ++n) {
    Ae[n] = -__expf(Alog[e * DST + n]);
    h[n]  = 0.f;
  }
  const float dp = Dp[e];
  for (int s = 0; s < SS; ++s) {
    const int t  = b * SS + s;
    const float dt  = dtb[(size_t)t * DI + e];
    const float xv  = xc[(size_t)t * DI + e];
    const float dbx = dt * xv;
    const float* Bt = Bm + (size_t)t * DST;
    const float* Ct = Cm + (size_t)t * DST;
    float y = 0.f;
#pragma unroll
    for (int n = 0; n < DST; ++n) {
      h[n] = __expf(dt * Ae[n]) * h[n] + dbx * Bt[n];
      y += h[n] * Ct[n];
    }
    const float zg = xz[(size_t)t * (2 * DI) + DI + e];
    yact[(size_t)t * DI + e] = (y + xv * dp) * siluf(zg);
  }
}

// ---------------------------------------------------------------------------
// LayerNorm over DM=128, one block (128 threads) per token.
// ---------------------------------------------------------------------------
__global__ __launch_bounds__(DM)
void k_ln(const float* __restrict__ m, const float* __restrict__ g,
          const float* __restrict__ bta, float* __restrict__ out) {
  __shared__ float red[4];
  const int t = blockIdx.x, f = threadIdx.x;
  const float v = m[(size_t)t * DM + f];
  float sum = v;
#pragma unroll
  for (int o = 16; o > 0; o >>= 1) sum += __shfl_xor(sum, o, 32);
  if ((f & 31) == 0) red[f >> 5] = sum;
  __syncthreads();
  const float mu = (red[0] + red[1] + red[2] + red[3]) * (1.f / DM);
  const float d  = v - mu;
  float sq = d * d;
#pragma unroll
  for (int o = 16; o > 0; o >>= 1) sq += __shfl_xor(sq, o, 32);
  __syncthreads();
  if ((f & 31) == 0) red[f >> 5] = sq;
  __syncthreads();
  const float var = (red[0] + red[1] + red[2] + red[3]) * (1.f / DM);
  out[(size_t)t * DM + f] = d * rsqrtf(var + 1e-5f) * g[f] + bta[f];
}

// ---------------------------------------------------------------------------
// z[b,s,t,f] = z0 + sum_i hist[i][b,s,f] * hist[i][b,t,f]
// Block: fixed (b, srow), 2 t-values x 128 f; 24x128 s-row vector in LDS.
// Single streaming write of the 269 MB output (the HBM roofline term).
// ---------------------------------------------------------------------------
__global__ __launch_bounds__(256)
void k_z(const float* __restrict__ z0, const float* __restrict__ hist,
         float* __restrict__ zout) {
  __shared__ float su[LYR * DM];   // 12 KB
  const int blk   = blockIdx.x;
  const int tpair = blk & (SS / 2 - 1);
  const int srow  = (blk >> 8) & (SS - 1);
  const int b     = blk >> 17;

  for (int i = threadIdx.x; i < LYR * DM; i += 256) {
    const int il = i >> 7, f = i & (DM - 1);
    su[i] = hist[((size_t)il * NT + (size_t)b * SS + srow) * DM + f];
  }
  __syncthreads();

  const int tl = threadIdx.x >> 7;          // 0..1
  const int f  = threadIdx.x & (DM - 1);
  const int t  = tpair * 2 + tl;
  const size_t oidx = ((((size_t)b * SS + srow) * SS) + t) * DM + f;
  float acc = z0[oidx];
  const float* hp = hist + ((size_t)b * SS + t) * DM + f;
#pragma unroll
  for (int i = 0; i < LYR; ++i)
    acc += su[i * DM + f] * hp[(size_t)i * NT * DM];
  zout[oidx] = acc;
}

// ---------------------------------------------------------------------------
extern "C" void kernel_launch(void* const* d_in, const int* in_sizes, int n_in,
                              void* d_out, int out_size, void* d_ws,
                              size_t ws_size, hipStream_t stream) {
  const float* s_in  = (const float*)d_in[0];
  const float* z0    = (const float*)d_in[1];
  const float* Wi_a  = (const float*)d_in[2];
  const float* cw_a  = (const float*)d_in[3];
  const float* cb_a  = (const float*)d_in[4];
  const float* Wx_a  = (const float*)d_in[5];
  const float* Wdt_a = (const float*)d_in[6];
  const float* bdt_a = (const float*)d_in[7];
  const float* Al_a  = (const float*)d_in[8];
  const float* Dp_a  = (const float*)d_in[9];
  const float* Wo_a  = (const float*)d_in[10];
  const float* lg_a  = (const float*)d_in[11];
  const float* lb_a  = (const float*)d_in[12];
  const float* lw_a  = (const float*)d_in[13];

  float* ws    = (float*)d_ws;
  float* s_cur = ws;  ws += (size_t)NT * DM;
  float* xz    = ws;  ws += (size_t)NT * 2 * DI;
  float* xc    = ws;  ws += (size_t)NT * DI;
  float* dtb   = ws;  ws += (size_t)NT * DI;
  float* Bmw   = ws;  ws += (size_t)NT * DST;
  float* Cmw   = ws;  ws += (size_t)NT * DST;
  float* yact  = ws;  ws += (size_t)NT * DI;
  float* mbuf  = ws;  ws += (size_t)NT * DM;
  float* mn    = ws;  ws += (size_t)NT * DM;
  float* hist  = ws;  // LYR * NT * DM floats (12.6 MB)

  hipMemcpyAsync(s_cur, s_in, (size_t)NT * DM * sizeof(float),
                 hipMemcpyDeviceToDevice, stream);

  for (int l = 0; l < LYR; ++l) {
    k_gemm4<DM, 2 * DI><<<dim3(NT / 16, (2 * DI) / 64), 32, 0, stream>>>(
        s_cur, Wi_a + (size_t)l * 2 * DI * DM, xz);
    k_conv_proj<<<NT, DI, 0, stream>>>(
        xz, cw_a + (size_t)l * DI * DCONV, cb_a + (size_t)l * DI,
        Wx_a + (size_t)l * (DTR + 2 * DST) * DI, Wdt_a + (size_t)l * DI * DTR,
        bdt_a + (size_t)l * DI, xc, dtb, Bmw, Cmw);
    k_scan<<<BB, DI, 0, stream>>>(xc, dtb, Bmw, Cmw,
                                  Al_a + (size_t)l * DI * DST,
                                  Dp_a + (size_t)l * DI, xz, yact);
    k_gemm4<DI, DM><<<dim3(NT / 16, DM / 64), 32, 0, stream>>>(
        yact, Wo_a + (size_t)l * DM * DI, mbuf);
    k_ln<<<NT, DM, 0, stream>>>(mbuf, lg_a + (size_t)l * DM,
                                lb_a + (size_t)l * DM, mn);
    k_lin_resid<<<dim3(NT / 16, DM / 64), 32, 0, stream>>>(
        mn, lw_a + (size_t)l * DM * DM, s_cur, hist + (size_t)l * NT * DM);
  }

  // outputs: s (131072 floats) then z (67108864 floats)
  hipMemcpyAsync(d_out, s_cur, (size_t)NT * DM * sizeof(float),
                 hipMemcpyDeviceToDevice, stream);
  float* zout = (float*)d_out + (size_t)NT * DM;
  k_z<<<BB * SS * (SS / 2), 256, 0, stream>>>(z0, hist, zout);
}